// MultiSubjectNIIRFPEFT_51144470560817
// MI455X (gfx1250) — compile-verified
//
#include <hip/hip_runtime.h>
#include <math.h>

typedef __attribute__((ext_vector_type(2))) float v2f;
typedef __attribute__((ext_vector_type(8))) float v8f;

#define NWAVES   4
#define XSTRIDE  132     // 128 + 4 dword pad -> conflict-free strided LDS access
#define HFDIM    128
#define NLOOK    1000

__constant__ float c_fc_low[6] = {20.f, 200.f, 500.f, 1000.f, 2000.f, 16000.f};
__constant__ float c_fc_wid[6] = {180.f, 300.f, 500.f, 1000.f, 2000.f, 6000.f};

__device__ __forceinline__ float gelu_exact(float x) {
    return 0.5f * x * (1.0f + erff(x * 0.70710678118654752440f));
}
__device__ __forceinline__ float sigm(float x) {
    return 1.0f / (1.0f + expf(-x));
}

extern "C" __global__ __launch_bounds__(NWAVES * 32, 1)
void niirf_peft_wmma(const float* __restrict__ phis,
                     const float* __restrict__ thetas,
                     const int*   __restrict__ sidxs,
                     const float* __restrict__ bmat,
                     const float* __restrict__ mlp_W,
                     const float* __restrict__ mlp_b,
                     const float* __restrict__ out_W,
                     const float* __restrict__ out_b,
                     const float* __restrict__ la_W,
                     const float* __restrict__ la_b,
                     const float* __restrict__ la_W_out,
                     const float* __restrict__ la_b_out,
                     const float* __restrict__ lb_W,
                     const float* __restrict__ lb_b,
                     float* __restrict__ out,
                     int batch)
{
    __shared__ float s_x[NWAVES][16 * XSTRIDE];
    __shared__ float s_mean[NWAVES][16];
    __shared__ int   s_sidx[NWAVES][16];

    const int tid  = threadIdx.x;
    const int wave = tid >> 5;
    const int lane = tid & 31;
    const int row  = lane & 15;   // sample row for A / feature col for B,D
    const int half = lane >> 4;

    const int base = (blockIdx.x * NWAVES + wave) * 16;
    if (base >= batch) return;    // uniform per wave

    float* xs = &s_x[wave][0];

    // ---- per-sample scalars (this lane's row) ----
    const float phi   = phis[base + row];
    const float theta = thetas[base + row];
    const int   sid   = sidxs[base + row];
    if (half == 0) s_sidx[wave][row] = sid;

    // ---- initial Fourier features: x[sm][0:64]=sin(emb), x[sm][64:128]=cos(emb) ----
    const float p0 = phi - 3.14159265358979323846f;
#pragma unroll
    for (int t = 0; t < 32; ++t) {
        const int j = half * 32 + t;
        const float e = p0 * bmat[2 * j + 0] + theta * bmat[2 * j + 1];
        xs[row * XSTRIDE + j]      = sinf(e);
        xs[row * XSTRIDE + j + 64] = cosf(e);
    }
    __builtin_amdgcn_wave_barrier();

    // ================= hidden layers (shared code body) =================
#pragma unroll 1
    for (int n = 0; n < 2; ++n) {
        // ---- mean(v * x) per sample ----
        const float* lbWr = lb_W + (size_t)n * NLOOK * HFDIM + (size_t)sid * HFDIM;
        const float* lbBr = lb_b + n * HFDIM;
        float part = 0.0f;
#pragma unroll
        for (int t = 0; t < 16; ++t) {
            const int j = half * 64 + t * 4;
            const float4 v4 = *(const float4*)(lbWr + j);
            const float4 b4 = *(const float4*)(lbBr + j);
            const float4 x4 = *(const float4*)(xs + row * XSTRIDE + j);
            part += (v4.x + b4.x) * x4.x + (v4.y + b4.y) * x4.y
                  + (v4.z + b4.z) * x4.z + (v4.w + b4.w) * x4.w;
        }
        const float tot = part + __shfl_xor(part, 16, 32);
        if (half == 0) s_mean[wave][row] = tot * (1.0f / 128.0f);
        __builtin_amdgcn_wave_barrier();

        // ---- A fragments for the full 16x128 block, kept in VGPRs ----
        v2f afr[32];
#pragma unroll
        for (int kk = 0; kk < 32; ++kk)
            afr[kk] = *(const v2f*)(xs + row * XSTRIDE + kk * 4 + half * 2);
        __builtin_amdgcn_wave_barrier();

        // ---- GEMM: Y = X * W^T, 8 N-tiles, K = 128 in steps of 4 ----
        const float* Wn = mlp_W + n * HFDIM * HFDIM;
        v8f acc[8];
#pragma unroll
        for (int nt = 0; nt < 8; ++nt) {
            v8f a = {0.f, 0.f, 0.f, 0.f, 0.f, 0.f, 0.f, 0.f};
            const float* wrow = Wn + (nt * 16 + row) * HFDIM + half * 2;
#pragma unroll
            for (int kk = 0; kk < 32; ++kk) {
                const v2f b = *(const v2f*)(wrow + kk * 4);
                a = __builtin_amdgcn_wmma_f32_16x16x4_f32(
                        false, afr[kk], false, b, (short)0, a, false, false);
            }
            acc[nt] = a;
        }
        __builtin_amdgcn_wave_barrier();

        // ---- epilogue: + bias + u*mean, gelu, write back to stage ----
        float mvv[8]; int sdd[8];
#pragma unroll
        for (int j = 0; j < 8; ++j) {
            mvv[j] = s_mean[wave][half * 8 + j];
            sdd[j] = s_sidx[wave][half * 8 + j];
        }
        const float* laWn = la_W + (size_t)n * NLOOK * HFDIM;
        const float* laBn = la_b + n * HFDIM;
#pragma unroll
        for (int nt = 0; nt < 8; ++nt) {
            const int f   = nt * 16 + row;
            const float mb  = mlp_b[n * HFDIM + f];
            const float lab = laBn[f];
#pragma unroll
            for (int j = 0; j < 8; ++j) {
                const int m = half * 8 + j;
                const float u = laWn[(size_t)sdd[j] * HFDIM + f] + lab;
                const float y = acc[nt][j] + mb + u * mvv[j];
                xs[m * XSTRIDE + f] = gelu_exact(y);
            }
        }
        __builtin_amdgcn_wave_barrier();
    }

    // ================= output layer (128 -> 32) =================
    {
        const float* lbWr = lb_W + (size_t)2 * NLOOK * HFDIM + (size_t)sid * HFDIM;
        const float* lbBr = lb_b + 2 * HFDIM;
        float part = 0.0f;
#pragma unroll
        for (int t = 0; t < 16; ++t) {
            const int j = half * 64 + t * 4;
            const float4 v4 = *(const float4*)(lbWr + j);
            const float4 b4 = *(const float4*)(lbBr + j);
            const float4 x4 = *(const float4*)(xs + row * XSTRIDE + j);
            part += (v4.x + b4.x) * x4.x + (v4.y + b4.y) * x4.y
                  + (v4.z + b4.z) * x4.z + (v4.w + b4.w) * x4.w;
        }
        const float tot = part + __shfl_xor(part, 16, 32);
        if (half == 0) s_mean[wave][row] = tot * (1.0f / 128.0f);
        __builtin_amdgcn_wave_barrier();

        v2f afr[32];
#pragma unroll
        for (int kk = 0; kk < 32; ++kk)
            afr[kk] = *(const v2f*)(xs + row * XSTRIDE + kk * 4 + half * 2);

        v8f acc[2];
#pragma unroll
        for (int nt = 0; nt < 2; ++nt) {
            v8f a = {0.f, 0.f, 0.f, 0.f, 0.f, 0.f, 0.f, 0.f};
            const float* wrow = out_W + (nt * 16 + row) * HFDIM + half * 2;
#pragma unroll
            for (int kk = 0; kk < 32; ++kk) {
                const v2f b = *(const v2f*)(wrow + kk * 4);
                a = __builtin_amdgcn_wmma_f32_16x16x4_f32(
                        false, afr[kk], false, b, (short)0, a, false, false);
            }
            acc[nt] = a;
        }
        __builtin_amdgcn_wave_barrier();

        float mvv[8]; int sdd[8];
#pragma unroll
        for (int j = 0; j < 8; ++j) {
            mvv[j] = s_mean[wave][half * 8 + j];
            sdd[j] = s_sidx[wave][half * 8 + j];
        }

        const int q = row;                               // 0..15 within group of 16
        const int qi = (q >= 6 && q < 12) ? (q - 6) : 0; // table index for fc path
        const float fcl = c_fc_low[qi];
        const float fcw = c_fc_wid[qi];
        const size_t fc_off = (size_t)batch * 12;
        const size_t fb_off = (size_t)batch * 24;

#pragma unroll
        for (int nt = 0; nt < 2; ++nt) {
            const int c = nt;              // est[..., c, q]
            const int f = c * 16 + q;
            const float ob  = out_b[f];
            const float lab = la_b_out[f];
#pragma unroll
            for (int j = 0; j < 8; ++j) {
                const int m  = half * 8 + j;
                const int gm = base + m;
                const float u = la_W_out[(size_t)sdd[j] * 32 + f] + lab;
                const float y = acc[nt][j] + ob + u * mvv[j];
                if (q < 6) {
                    out[(size_t)(gm * 2 + c) * 6 + q] = y;                       // gain
                } else if (q < 12) {
                    out[fc_off + (size_t)(gm * 2 + c) * 6 + (q - 6)]
                        = fcw * sigm(y) + fcl;                                   // fc
                } else {
                    out[fb_off + (size_t)(gm * 2 + c) * 4 + (q - 12)]
                        = 3900.0f * sigm(y) + 100.0f;                            // fb
                }
            }
        }
    }
}

extern "C" void kernel_launch(void* const* d_in, const int* in_sizes, int n_in,
                              void* d_out, int out_size, void* d_ws, size_t ws_size,
                              hipStream_t stream) {
    (void)n_in; (void)out_size; (void)d_ws; (void)ws_size;
    const float* phis     = (const float*)d_in[0];
    const float* thetas   = (const float*)d_in[1];
    const int*   sidxs    = (const int*)  d_in[2];
    const float* bmat     = (const float*)d_in[3];
    const float* mlp_W    = (const float*)d_in[4];
    const float* mlp_b    = (const float*)d_in[5];
    const float* out_W    = (const float*)d_in[6];
    const float* out_b    = (const float*)d_in[7];
    const float* la_W     = (const float*)d_in[8];
    const float* la_b     = (const float*)d_in[9];
    const float* la_W_out = (const float*)d_in[10];
    const float* la_b_out = (const float*)d_in[11];
    const float* lb_W     = (const float*)d_in[12];
    const float* lb_b     = (const float*)d_in[13];
    float* out = (float*)d_out;

    const int batch  = in_sizes[0];
    const int blocks = (batch + NWAVES * 16 - 1) / (NWAVES * 16);

    hipLaunchKernelGGL(niirf_peft_wmma, dim3(blocks), dim3(NWAVES * 32), 0, stream,
                       phis, thetas, sidxs, bmat, mlp_W, mlp_b, out_W, out_b,
                       la_W, la_b, la_W_out, la_b_out, lb_W, lb_b, out, batch);
}